// Rotation_1151051235360
// MI455X (gfx1250) — compile-verified
//
#include <hip/hip_runtime.h>

typedef float v2f __attribute__((ext_vector_type(2)));
typedef float v8f __attribute__((ext_vector_type(8)));

#define DIM_TOT   5184
#define FREQ_OFF0 1536
#define J16_OFF   1696

__host__ __device__ constexpr int k4_of(int l) { return (2 * l + 4) / 4; }
__host__ __device__ constexpr int ap_off(int l) {
  int o = 0;
  for (int t = 1; t < l; ++t) o += k4_of(t) * 64;
  return o;
}

// ---------------------------------------------------------------------------
// Setup kernel: computes, per l = 1..8 (one wave each, matrices in LDS):
//   - real SO3 generators Xx, Xy in the real-SH basis (complex q-conjugation)
//   - freq table  freq[i] = Xy[i][d-1-i]            -> ws[FREQ_OFF0 + (l-1)*20]
//   - J = expm(pi/sqrt(2) * (Xx + Xy))  (scale-and-square, 16-term Taylor)
//   - J packed into the WMMA A-operand register layout -> ws[ap_off(l)]
//   - l==8: row 16 of J as scalar coefficients       -> ws[J16_OFF]
// Deterministic; re-run every kernel_launch (no cross-call state).
// ---------------------------------------------------------------------------
__global__ void __launch_bounds__(256) wigner_setup(float* __restrict__ ws) {
  __shared__ float pool[8][6][289];
  const int wv = threadIdx.x >> 5;
  const int lane = threadIdx.x & 31;
  const int l = wv + 1;
  const int d = 2 * l + 1;
  float* QR = pool[wv][0];
  float* QI = pool[wv][1];
  float* R  = pool[wv][2];
  float* TM = pool[wv][3];
  float* XA = pool[wv][4];
  float* SS = pool[wv][5];

  for (int e = lane; e < 289; e += 32) {
    QR[e] = 0.f; QI[e] = 0.f; R[e] = 0.f; TM[e] = 0.f; XA[e] = 0.f; SS[e] = 0.f;
  }
  __syncthreads();

  if (lane == 0) {
    const float is2 = 0.7071067811865475f;
    for (int mm = -l; mm < 0; ++mm) {
      int row = l + mm;
      QR[row * 17 + (l - mm)] = is2;     // 1/sqrt2
      QI[row * 17 + (l + mm)] = -is2;    // -i/sqrt2
    }
    QR[l * 17 + l] = 1.f;
    for (int mm = 1; mm <= l; ++mm) {
      int row = l + mm;
      float sg = (mm & 1) ? -1.f : 1.f;
      QR[row * 17 + (l + mm)] = sg * is2;
      QI[row * 17 + (l - mm)] = sg * is2; // i*(-1)^mm/sqrt2
    }
    int ph = l & 3;  // multiply q by (-i)^l
    if (ph) {
      for (int e = 0; e < d * 17; ++e) {
        float a = QR[e], b2 = QI[e];
        if (ph == 1)      { QR[e] =  b2; QI[e] = -a;  }   // * -i
        else if (ph == 2) { QR[e] = -a;  QI[e] = -b2; }   // * -1
        else              { QR[e] = -b2; QI[e] =  a;  }   // *  i
      }
    }
    // R = 0.5*(raising + lowering)
    for (int i = 0; i < d - 1; ++i) {
      float m1 = (float)(-l + i);
      R[(i + 1) * 17 + i] = -0.5f * sqrtf((float)(l * (l + 1)) - m1 * (m1 + 1.f));
      float m2 = (float)(-l + 1 + i);
      R[i * 17 + (i + 1)] = 0.5f * sqrtf((float)(l * (l + 1)) - m2 * (m2 - 1.f));
    }
  }
  __syncthreads();

  // TM = R * QR
  for (int e = lane; e < d * d; e += 32) {
    int i = e / d, j = e % d; float s = 0.f;
    for (int k = 0; k < d; ++k) s += R[i * 17 + k] * QR[k * 17 + j];
    TM[i * 17 + j] = s;
  }
  __syncthreads();
  // XA = QR^T * TM   (first half of Re(q^H * R * q))
  for (int e = lane; e < d * d; e += 32) {
    int i = e / d, j = e % d; float s = 0.f;
    for (int k = 0; k < d; ++k) s += QR[k * 17 + i] * TM[k * 17 + j];
    XA[i * 17 + j] = s;
  }
  __syncthreads();
  // TM = R * QI
  for (int e = lane; e < d * d; e += 32) {
    int i = e / d, j = e % d; float s = 0.f;
    for (int k = 0; k < d; ++k) s += R[i * 17 + k] * QI[k * 17 + j];
    TM[i * 17 + j] = s;
  }
  __syncthreads();
  // XA += QI^T * TM   -> XA = real Xx
  for (int e = lane; e < d * d; e += 32) {
    int i = e / d, j = e % d; float s = 0.f;
    for (int k = 0; k < d; ++k) s += QI[k * 17 + i] * TM[k * 17 + j];
    XA[i * 17 + j] += s;
  }
  __syncthreads();
  // TM = real Xy = Qi^T D Qr - Qr^T D Qi, D = diag(m)
  for (int e = lane; e < d * d; e += 32) {
    int i = e / d, j = e % d; float s = 0.f;
    for (int k = 0; k < d; ++k) {
      float mk = (float)(k - l);
      s += mk * (QI[k * 17 + i] * QR[k * 17 + j] - QR[k * 17 + i] * QI[k * 17 + j]);
    }
    TM[i * 17 + j] = s;
  }
  __syncthreads();
  // freq table (padded to 20 with zeros)
  for (int jj = lane; jj < 20; jj += 32)
    ws[FREQ_OFF0 + (l - 1) * 20 + jj] = (jj < d) ? TM[jj * 17 + (d - 1 - jj)] : 0.f;
  // XA = (pi/sqrt2)/64 * (Xx + Xy)   (already divided by 2^6 for expm)
  const float cc = 3.14159265358979323846f * 0.7071067811865475f * (1.f / 64.f);
  for (int e = lane; e < d * d; e += 32) {
    int i = e / d, j = e % d;
    XA[i * 17 + j] = cc * (XA[i * 17 + j] + TM[i * 17 + j]);
  }
  __syncthreads();
  // expm: SS = I + XA; T(QR) = XA; Taylor to n=16
  for (int e = lane; e < d * d; e += 32) {
    int i = e / d, j = e % d;
    QR[i * 17 + j] = XA[i * 17 + j];
    SS[i * 17 + j] = ((i == j) ? 1.f : 0.f) + XA[i * 17 + j];
  }
  __syncthreads();
  for (int n = 2; n <= 16; ++n) {
    float rn = 1.f / (float)n;
    for (int e = lane; e < d * d; e += 32) {
      int i = e / d, j = e % d; float s = 0.f;
      for (int k = 0; k < d; ++k) s += QR[i * 17 + k] * XA[k * 17 + j];
      TM[i * 17 + j] = s * rn;
    }
    __syncthreads();
    for (int e = lane; e < d * d; e += 32) {
      int i = e / d, j = e % d;
      SS[i * 17 + j] += TM[i * 17 + j];
      QR[i * 17 + j] = TM[i * 17 + j];
    }
    __syncthreads();
  }
  // 6 squarings
  for (int t = 0; t < 6; ++t) {
    for (int e = lane; e < d * d; e += 32) {
      int i = e / d, j = e % d; float s = 0.f;
      for (int k = 0; k < d; ++k) s += SS[i * 17 + k] * SS[k * 17 + j];
      QI[i * 17 + j] = s;
    }
    __syncthreads();
    for (int e = lane; e < d * d; e += 32) {
      int i = e / d, j = e % d;
      SS[i * 17 + j] = QI[i * 17 + j];
    }
    __syncthreads();
  }
  // Pack J into WMMA f32 16x16x4 A-operand layout:
  //  A[M=lane%16][K = 4*kc + v + 2*(lane>>4)], v interleaved per lane (b64 loads)
  const int K4 = k4_of(l);
  const int ao = ap_off(l);
  for (int kc = 0; kc < K4; ++kc) {
    int i = lane & 15;
    int kb = 4 * kc + 2 * (lane >> 4);
    for (int v = 0; v < 2; ++v) {
      int k2 = kb + v;
      ws[ao + kc * 64 + 2 * lane + v] = (i < d && k2 < d) ? SS[i * 17 + k2] : 0.f;
    }
  }
  if (l == 8) {
    for (int jj = lane; jj < 20; jj += 32)
      ws[J16_OFF + jj] = (jj < 17) ? SS[16 * 17 + jj] : 0.f;
  }
}

// ---------------------------------------------------------------------------
// Dy(theta) applied in WMMA D-register layout: vgpr r, lanes0-15 hold row i=r,
// lanes16-31 row i=r+8.  Partner row (d-1-i) reachable with lane^16 shuffles.
// ---------------------------------------------------------------------------
template <int L>
__device__ __forceinline__ void dy_stage(float (&M)[8], float& r16, float th,
                                         const float (&fD)[8], float f16) {
  constexpr int D = 2 * L + 1;
  const bool lo = ((threadIdx.x & 16) == 0);
  float old[8], osw[8];
#pragma unroll
  for (int r = 0; r < 8; ++r) old[r] = M[r];
#pragma unroll
  for (int r = 0; r < 8; ++r) osw[r] = __shfl_xor(old[r], 16, 32);
  float o16 = r16;
#pragma unroll
  for (int r = 0; r < 8; ++r) {
    int p0 = D - 1 - r;  // mirror row for lanes0-15 (i=r)
    int p1 = D - 9 - r;  // mirror row for lanes16-31 (i=r+8)
    float s0 = 0.f, s1 = 0.f;
    if (p0 >= 0 && p0 < D) s0 = (p0 == 16) ? o16 : ((p0 < 8) ? old[p0] : osw[p0 - 8]);
    if (p1 >= 0 && p1 < D) s1 = (p1 >= 8) ? old[p1 - 8] : osw[p1];
    float P = lo ? s0 : s1;
    float sn, cs;
    __sincosf(fD[r] * th, &sn, &cs);
    M[r] = cs * old[r] + sn * P;
  }
  if constexpr (L == 8) {  // extra row 16 (valid in lanes0-15): mirror is row 0
    float sn, cs;
    __sincosf(f16 * th, &sn, &cs);
    r16 = cs * o16 + sn * old[0];
  }
}

// ---------------------------------------------------------------------------
// One (l, 16-column tile) unit; wave iterates batch elements b = grp + e*G.
// Pipeline: load seg tile fused with Dy(gamma) -> WMMA J -> Dy(beta)
//           -> half-swap repack -> WMMA J -> Dy(alpha) -> store.
// ---------------------------------------------------------------------------
template <int L>
__device__ void process_l(const float* __restrict__ x, float* __restrict__ out,
                          const float* __restrict__ gam, const float* __restrict__ bet,
                          const float* __restrict__ alp, const float* __restrict__ ws,
                          int tile, int grp, int G, int nb) {
  constexpr int D = 2 * L + 1;
  constexpr int K4 = (2 * L + 4) / 4;
  constexpr int LOFF = 64 * L * L;
  const int lane = threadIdx.x & 31;
  const int mcol = lane & 15;
  const int half = lane >> 4;
  const bool lo = (half == 0);
  const int c0 = tile * 16;

  // Constant A operand (J) — packed layout, one b64 load per chunk.
  const float* __restrict__ ap = ws + ap_off(L);
  v2f Areg[K4];
#pragma unroll
  for (int kc = 0; kc < K4; ++kc) Areg[kc] = ((const v2f*)ap)[kc * 32 + lane];

  const float* __restrict__ ftab = ws + FREQ_OFF0 + (L - 1) * 20;
  int offj[K4][2], offm[K4][2];
  float fB[K4][2], vmask[K4][2];
#pragma unroll
  for (int kc = 0; kc < K4; ++kc)
#pragma unroll
    for (int v = 0; v < 2; ++v) {
      int j = 4 * kc + v + 2 * half;  // B-operand row (K index) held by this lane
      bool ok = (j < D);
      int jc = ok ? j : 0;
      offj[kc][v] = 64 * jc + c0 + mcol;
      offm[kc][v] = 64 * (D - 1 - jc) + c0 + mcol;
      fB[kc][v] = ftab[j];            // zero-padded beyond D
      vmask[kc][v] = ok ? 1.f : 0.f;
    }
  float fD[8];
#pragma unroll
  for (int r = 0; r < 8; ++r) fD[r] = ftab[r + 8 * half];
  float f16 = 0.f;
  float j16c[17] = {};
  if constexpr (L == 8) {
    f16 = ftab[16];
    const float* jj = ws + J16_OFF;
#pragma unroll
    for (int j = 0; j < 17; ++j) j16c[j] = jj[j];
  }

  for (int b = grp; b < nb; b += G) {
    const size_t ebase = (size_t)b * DIM_TOT + LOFF;
    const float ga = gam[b], be = bet[b], al = alp[b];

    // B1 = Dy(gamma) applied at load (row + mirror-row gathers, mirror hits L0)
    float B1v[K4][2];
#pragma unroll
    for (int kc = 0; kc < K4; ++kc)
#pragma unroll
      for (int v = 0; v < 2; ++v) {
        float a = x[ebase + offj[kc][v]];
        float m_ = x[ebase + offm[kc][v]];
        float sn, cs;
        __sincosf(fB[kc][v] * ga, &sn, &cs);
        B1v[kc][v] = vmask[kc][v] * (cs * a + sn * m_);
      }

    // l=8: row 16 of J*B1 via scalar-coefficient VALU dot
    float r16 = 0.f;
    if constexpr (L == 8) {
#pragma unroll
      for (int j = 0; j < 17; ++j) {
        float val = B1v[j >> 2][j & 1];
        if ((j >> 1) & 1) val = __shfl_xor(val, 16, 32);
        r16 = fmaf(j16c[j], val, r16);
      }
    }

    // chain 1: D1 = J x B1
    v8f acc{0.f, 0.f, 0.f, 0.f, 0.f, 0.f, 0.f, 0.f};
#pragma unroll
    for (int kc = 0; kc < K4; ++kc) {
      v2f bb; bb.x = B1v[kc][0]; bb.y = B1v[kc][1];
      acc = __builtin_amdgcn_wmma_f32_16x16x4_f32(false, Areg[kc], false, bb,
                                                  (short)0, acc, false, false);
    }
    float M[8];
#pragma unroll
    for (int r = 0; r < 8; ++r) M[r] = acc[r];

    dy_stage<L>(M, r16, be, fD, f16);

    // repack D layout -> B layout (lane^16 swaps + register renames)
    float msw[8];
#pragma unroll
    for (int r = 0; r < 8; ++r) msw[r] = __shfl_xor(M[r], 16, 32);

    float B2v[K4][2];
#pragma unroll
    for (int kc = 0; kc < K4; ++kc)
#pragma unroll
      for (int v = 0; v < 2; ++v) {
        int j0 = 4 * kc + v, j1 = j0 + 2;
        float s0 = 0.f, s1 = 0.f;
        if (j0 < D) s0 = (j0 == 16) ? r16 : ((j0 < 8) ? M[j0] : msw[j0 - 8]);
        if (j1 < D) s1 = (j1 >= 8) ? M[j1 - 8] : msw[j1];
        B2v[kc][v] = lo ? s0 : s1;
      }

    float r16b = 0.f;
    if constexpr (L == 8) {
#pragma unroll
      for (int j = 0; j < 17; ++j) {
        float val = (j == 16) ? r16 : ((j < 8) ? M[j] : msw[j - 8]);
        r16b = fmaf(j16c[j], val, r16b);
      }
    }

    // chain 2: D2 = J x B2
    v8f acc2{0.f, 0.f, 0.f, 0.f, 0.f, 0.f, 0.f, 0.f};
#pragma unroll
    for (int kc = 0; kc < K4; ++kc) {
      v2f bb; bb.x = B2v[kc][0]; bb.y = B2v[kc][1];
      acc2 = __builtin_amdgcn_wmma_f32_16x16x4_f32(false, Areg[kc], false, bb,
                                                   (short)0, acc2, false, false);
    }
    float O[8];
#pragma unroll
    for (int r = 0; r < 8; ++r) O[r] = acc2[r];

    dy_stage<L>(O, r16b, al, fD, f16);

#pragma unroll
    for (int r = 0; r < 8; ++r) {
      int i = r + 8 * half;
      if (i < D) out[ebase + 64 * i + c0 + mcol] = O[r];
    }
    if constexpr (L == 8) {
      if (lo) out[ebase + 64 * 16 + c0 + mcol] = r16b;
    }
  }
}

// Wave w in each 16-wave group handles the pair (l, 9-l) x one column tile:
// per-tile WMMA-chunk costs [2,4,4,6,6,8,8,10] pair to exactly 12 -> balanced.
__global__ void __launch_bounds__(256) rot_kernel(
    const float* __restrict__ gam, const float* __restrict__ bet,
    const float* __restrict__ alp, const float* __restrict__ x,
    float* __restrict__ out, const float* __restrict__ ws, int nb, int G) {
  const int wave = (int)((blockIdx.x * blockDim.x + threadIdx.x) >> 5);
  const int grp = wave >> 4;
  const int w = wave & 15;
  if (grp >= G) return;
  const int p = w >> 2;
  const int tile = w & 3;
  switch (p) {
    case 0:
      process_l<1>(x, out, gam, bet, alp, ws, tile, grp, G, nb);
      process_l<8>(x, out, gam, bet, alp, ws, tile, grp, G, nb);
      break;
    case 1:
      process_l<2>(x, out, gam, bet, alp, ws, tile, grp, G, nb);
      process_l<7>(x, out, gam, bet, alp, ws, tile, grp, G, nb);
      break;
    case 2:
      process_l<3>(x, out, gam, bet, alp, ws, tile, grp, G, nb);
      process_l<6>(x, out, gam, bet, alp, ws, tile, grp, G, nb);
      break;
    default:
      process_l<4>(x, out, gam, bet, alp, ws, tile, grp, G, nb);
      process_l<5>(x, out, gam, bet, alp, ws, tile, grp, G, nb);
      break;
  }
}

// l = 0: identity (freq=0, J=[[1]]) -> straight copy of first 64 floats/elem.
__global__ void __launch_bounds__(256) l0_copy(const float* __restrict__ x,
                                               float* __restrict__ out, int nb) {
  int t = blockIdx.x * blockDim.x + threadIdx.x;
  int b = t >> 6, m = t & 63;
  if (b < nb) out[(size_t)b * DIM_TOT + m] = x[(size_t)b * DIM_TOT + m];
}

extern "C" void kernel_launch(void* const* d_in, const int* in_sizes, int n_in,
                              void* d_out, int out_size, void* d_ws, size_t ws_size,
                              hipStream_t stream) {
  const float* gam = (const float*)d_in[0];
  const float* bet = (const float*)d_in[1];
  const float* alp = (const float*)d_in[2];
  const float* x = (const float*)d_in[3];
  float* out = (float*)d_out;
  float* ws = (float*)d_ws;
  const int nb = in_sizes[0];

  wigner_setup<<<1, 256, 0, stream>>>(ws);
  l0_copy<<<(nb * 64 + 255) / 256, 256, 0, stream>>>(x, out, nb);
  int G = nb < 1024 ? nb : 1024;
  rot_kernel<<<2 * G, 256, 0, stream>>>(gam, bet, alp, x, out, ws, nb, G);
}